// BiRNNLM_7842610282760
// MI455X (gfx1250) — compile-verified
//
#include <hip/hip_runtime.h>

#define TT 1024
#define BB 8
#define EE 32
#define HH 8
#define VV 32000

typedef __attribute__((ext_vector_type(2))) float v2f;
typedef __attribute__((ext_vector_type(8))) float v8f;

// ---------------------------------------------------------------------------
// Kernel 0: xproj[dir][t][b][j] = emb[x[tt][b]] @ Wx + bx[j] + bh[j]
// ---------------------------------------------------------------------------
__global__ __launch_bounds__(256) void k_xproj(
    const int* __restrict__ x, const float* __restrict__ emb,
    const float* __restrict__ Wx1, const float* __restrict__ bx1, const float* __restrict__ bh1,
    const float* __restrict__ Wx2, const float* __restrict__ bx2, const float* __restrict__ bh2,
    float* __restrict__ xproj)
{
    int idx = blockIdx.x * 256 + threadIdx.x;          // 131072 total
    int j   = idx & 7;
    int b   = (idx >> 3) & 7;
    int t   = (idx >> 6) & 1023;
    int dir = idx >> 16;
    int tt  = dir ? (TT - 1 - t) : t;

    int tok = x[tt * BB + b];
    const float* Wx = dir ? Wx2 : Wx1;
    float acc = dir ? (bx2[j] + bh2[j]) : (bx1[j] + bh1[j]);
    const float* er = emb + (size_t)tok * EE;
#pragma unroll
    for (int e = 0; e < EE; e++)
        acc = __builtin_fmaf(er[e], Wx[e * HH + j], acc);
    xproj[idx] = acc;
}

// ---------------------------------------------------------------------------
// Kernel 1: serial bidirectional scan, ONE wave32, zero barriers.
// lane = dir*16 + b*2 + half ; each lane owns h[half*4 .. half*4+3] of (dir,b).
// Partner half exchanged with ds_swizzle SWAPX1 (xor=1): offset = 0x041F.
// Emits h BEFORE update (matches reference scan): t for dir0, T-1-s for dir1.
// totalh layout: [T*B rows][16] = [h1(8) | h2(8)]
// ---------------------------------------------------------------------------
__device__ __forceinline__ float swap1(float v) {
    return __int_as_float(__builtin_amdgcn_ds_swizzle(__float_as_int(v), 0x041F));
}

__global__ __launch_bounds__(32) void k_scan(
    const float* __restrict__ Wh1, const float* __restrict__ Wh2,
    const float* __restrict__ xproj, float* __restrict__ totalh)
{
    int lane = threadIdx.x;
    int dir  = lane >> 4;
    int b    = (lane >> 1) & 7;
    int half = lane & 1;

    const float* Wh = dir ? Wh2 : Wh1;
    float w[8][4];
#pragma unroll
    for (int k = 0; k < 8; k++)
#pragma unroll
        for (int j = 0; j < 4; j++)
            w[k][j] = Wh[k * HH + half * 4 + j];

    float h[4] = {0.f, 0.f, 0.f, 0.f};
    const float* xp = xproj + dir * (TT * BB * HH) + b * HH + half * 4;
    const float L2E = 1.4426950408889634f;

    for (int s = 0; s < TT; s++) {
        int t = dir ? (TT - 1 - s) : s;
        float4 st; st.x = h[0]; st.y = h[1]; st.z = h[2]; st.w = h[3];
        *reinterpret_cast<float4*>(totalh + (size_t)(t * BB + b) * 16 + dir * 8 + half * 4) = st;

        float4 xv = *reinterpret_cast<const float4*>(xp + s * (BB * HH));
        float xa[4] = {xv.x, xv.y, xv.z, xv.w};

        float ho[4];
#pragma unroll
        for (int i = 0; i < 4; i++) ho[i] = swap1(h[i]);

        float hk[8];
#pragma unroll
        for (int i = 0; i < 4; i++) {           // branch-free half merge
            hk[i]     = half ? ho[i] : h[i];
            hk[4 + i] = half ? h[i]  : ho[i];
        }
#pragma unroll
        for (int j = 0; j < 4; j++) {
            float a = xa[j];
#pragma unroll
            for (int k = 0; k < 8; k++)
                a = __builtin_fmaf(hk[k], w[k][j], a);
            float e = __builtin_amdgcn_exp2f(-a * L2E);     // sigmoid
            h[j] = __builtin_amdgcn_rcpf(1.0f + e);
        }
    }
}

// ---------------------------------------------------------------------------
// Kernel 2: logits + log_softmax via V_WMMA_F32_16X16X4_F32, two passes.
// One workgroup per 16 rows; 8 waves each own 250 of the 2000 V-tiles
// (compile-time-constant trip count -> uniform loop, no exec masking).
// Pass1: accumulate sum(exp(logit)). Pass2: recompute, NT-store normalized.
// ---------------------------------------------------------------------------
__global__ __launch_bounds__(256) void k_logits(
    const float* __restrict__ totalh, const float* __restrict__ Wout,
    float* __restrict__ out)
{
    const int tid  = threadIdx.x;
    const int wave = tid >> 5;
    const int lane = tid & 31;
    const int m    = lane & 15;          // A row / B column / C column
    const int kh   = lane >> 4;          // K-half select
    const int rowbase = blockIdx.x * 16;

    __shared__ float TH[16 * 16];
    __shared__ float red[8][16];
    __shared__ float rowc[16];

    TH[tid] = totalh[(size_t)rowbase * 16 + tid];
    __syncthreads();

    // A fragments: 16x4 f32 tile i covers K = 4i..4i+3
    v2f a[4];
#pragma unroll
    for (int i = 0; i < 4; i++) {
        a[i].x = TH[m * 16 + 4 * i + 2 * kh];
        a[i].y = TH[m * 16 + 4 * i + 2 * kh + 1];
    }

    const float L2E = 1.4426950408889634f;
    float s[8] = {0.f, 0.f, 0.f, 0.f, 0.f, 0.f, 0.f, 0.f};

    // -------- pass 1: sum of exp(logit) per row --------
#pragma unroll 2
    for (int it = 0; it < (VV / 16) / 8; it++) {
        const int vb = (it * 8 + wave) * 16 + m;
        v8f c = {0.f, 0.f, 0.f, 0.f, 0.f, 0.f, 0.f, 0.f};
#pragma unroll
        for (int i = 0; i < 4; i++) {
            v2f bf;
            bf.x = Wout[(size_t)(4 * i + 2 * kh    ) * VV + vb];
            bf.y = Wout[(size_t)(4 * i + 2 * kh + 1) * VV + vb];
            c = __builtin_amdgcn_wmma_f32_16x16x4_f32(
                    false, a[i], false, bf, (short)0, c, false, false);
        }
#pragma unroll
        for (int r = 0; r < 8; r++)
            s[r] += __builtin_amdgcn_exp2f(c[r] * L2E);
    }

    // reduce across the 16 lanes of each half (rows 0-7 in lanes 0-15, 8-15 in 16-31)
#pragma unroll
    for (int r = 0; r < 8; r++) {
#pragma unroll
        for (int mask = 1; mask < 16; mask <<= 1)
            s[r] += __shfl_xor(s[r], mask, 32);
    }
    if (lane == 0 || lane == 16) {
#pragma unroll
        for (int r = 0; r < 8; r++) red[wave][kh * 8 + r] = s[r];
    }
    __syncthreads();
    if (tid < 16) {
        float tot = 0.f;
#pragma unroll
        for (int w = 0; w < 8; w++) tot += red[w][tid];
        rowc[tid] = __builtin_amdgcn_logf(tot) * 0.69314718055994531f;  // ln(sum)
    }
    __syncthreads();

    float rc[8];
#pragma unroll
    for (int r = 0; r < 8; r++) rc[r] = rowc[kh * 8 + r];

    // -------- pass 2: recompute logits, NT-store normalized --------
#pragma unroll 2
    for (int it = 0; it < (VV / 16) / 8; it++) {
        const int vb = (it * 8 + wave) * 16 + m;
        v8f c = {0.f, 0.f, 0.f, 0.f, 0.f, 0.f, 0.f, 0.f};
#pragma unroll
        for (int i = 0; i < 4; i++) {
            v2f bf;
            bf.x = Wout[(size_t)(4 * i + 2 * kh    ) * VV + vb];
            bf.y = Wout[(size_t)(4 * i + 2 * kh + 1) * VV + vb];
            c = __builtin_amdgcn_wmma_f32_16x16x4_f32(
                    false, a[i], false, bf, (short)0, c, false, false);
        }
#pragma unroll
        for (int r = 0; r < 8; r++)
            __builtin_nontemporal_store(
                c[r] - rc[r],
                out + (size_t)(rowbase + kh * 8 + r) * VV + vb);
    }
}

// ---------------------------------------------------------------------------
extern "C" void kernel_launch(void* const* d_in, const int* in_sizes, int n_in,
                              void* d_out, int out_size, void* d_ws, size_t ws_size,
                              hipStream_t stream)
{
    const int*   x    = (const int*)  d_in[0];
    const float* emb  = (const float*)d_in[1];
    const float* Wx1  = (const float*)d_in[2];
    const float* bx1  = (const float*)d_in[3];
    const float* Wh1  = (const float*)d_in[4];
    const float* bh1  = (const float*)d_in[5];
    const float* Wx2  = (const float*)d_in[6];
    const float* bx2  = (const float*)d_in[7];
    const float* Wh2  = (const float*)d_in[8];
    const float* bh2  = (const float*)d_in[9];
    const float* Wout = (const float*)d_in[10];
    float* out = (float*)d_out;

    float* xproj  = (float*)d_ws;                    // 2*T*B*H = 131072 floats
    float* totalh = xproj + 2 * TT * BB * HH;        // T*B*16  = 131072 floats

    k_xproj <<<512, 256, 0, stream>>>(x, emb, Wx1, bx1, bh1, Wx2, bx2, bh2, xproj);
    k_scan  <<<1,   32,  0, stream>>>(Wh1, Wh2, xproj, totalh);
    k_logits<<<512, 256, 0, stream>>>(totalh, Wout, out);
}